// FeatureLoss_11759620456547
// MI455X (gfx1250) — compile-verified
//
#include <hip/hip_runtime.h>
#include <hip/hip_bf16.h>
#include <math.h>

// ---- problem constants (fixed by reference) ----
#define NBATCH 8
#define CCH    256
#define HDIM   128
#define WDIM   128
#define HW     16384   // HDIM*WDIM
#define NB     20
// TEMP = 0.5 ; fea logit = (fea_sum/C)/TEMP = fea_sum/128 ; ch logit = ch_sum/8192

typedef float v2f __attribute__((ext_vector_type(2)));
typedef float v8f __attribute__((ext_vector_type(8)));

// ---- workspace layout (float offsets) ----
#define OFF_FEA_S  0u          // [N,HW]   sum_c |s|
#define OFF_FEA_T  131072u     // [N,HW]
#define OFF_CM_S   262144u     // [N,HW]   spatial-pool logits S
#define OFF_CM_T   393216u     // [N,HW]
#define OFF_SM_S   524288u     // [N,HW]   softmax(cm_s)
#define OFF_SM_T   655360u     // [N,HW]
#define OFF_WFG    786432u     // [N,HW]   S_t_att * Mask_fg
#define OFF_WBG    917504u     // [N,HW]   S_t_att * Mask_bg
#define OFF_CH_S   1048576u    // [N,C]    sum_hw |s|
#define OFF_CH_T   1050624u
#define OFF_CTA    1052672u    // [N,C]    C_t attention
#define OFF_CTX_S  1054720u    // [N,C]
#define OFF_CTX_T  1056768u
#define OFF_SD     1058816u    // [N,C]  sum_hw (s-t)
#define OFF_SS2    1060864u    // [N,C]  sum_hw (s-t)^2
#define OFF_SPAR   1062912u    // [4][N][2] softmax {max,sumexp} for feaS,feaT,cmS,cmT
#define OFF_ACC    1062976u    // [4] accumulators: fg, bg, chL1, spL1

__device__ __forceinline__ float block_reduce_sum(float v, float* red) {
    int tid = threadIdx.x;
    red[tid] = v; __syncthreads();
    for (int s = 128; s > 0; s >>= 1) { if (tid < s) red[tid] += red[tid + s]; __syncthreads(); }
    float r = red[0]; __syncthreads(); return r;
}
__device__ __forceinline__ float block_reduce_max(float v, float* red) {
    int tid = threadIdx.x;
    red[tid] = v; __syncthreads();
    for (int s = 128; s > 0; s >>= 1) { if (tid < s) red[tid] = fmaxf(red[tid], red[tid + s]); __syncthreads(); }
    float r = red[0]; __syncthreads(); return r;
}

// ---------------- K0: zero atomic accumulators ----------------
__global__ void k_init(float* acc) {
    if (threadIdx.x < 4) acc[threadIdx.x] = 0.f;
}

// ---------------- K1a: per-pixel sums (fea, cm) ----------------
// grid (16, 8): blockIdx.x = hw chunk of 1024, blockIdx.y = n. 256 thr, float4 each.
__global__ __launch_bounds__(256) void k_pass1a(
    const float* __restrict__ S, const float* __restrict__ T,
    const float* __restrict__ wms, const float* __restrict__ wmt,
    float* __restrict__ feaS, float* __restrict__ feaT,
    float* __restrict__ cmS, float* __restrict__ cmT)
{
    int n = blockIdx.y;
    int hw0 = blockIdx.x * 1024 + threadIdx.x * 4;
    float4 fS = {0,0,0,0}, fT = {0,0,0,0}, cS = {0,0,0,0}, cT = {0,0,0,0};
    size_t base = (size_t)n * CCH * HW + hw0;
    #pragma unroll 4
    for (int c = 0; c < CCH; ++c) {
        float4 s = *(const float4*)(S + base + (size_t)c * HW);
        float4 t = *(const float4*)(T + base + (size_t)c * HW);
        float a = wms[c], b = wmt[c];
        fS.x += fabsf(s.x); fS.y += fabsf(s.y); fS.z += fabsf(s.z); fS.w += fabsf(s.w);
        fT.x += fabsf(t.x); fT.y += fabsf(t.y); fT.z += fabsf(t.z); fT.w += fabsf(t.w);
        cS.x += s.x*a; cS.y += s.y*a; cS.z += s.z*a; cS.w += s.w*a;
        cT.x += t.x*b; cT.y += t.y*b; cT.z += t.z*b; cT.w += t.w*b;
    }
    size_t o = (size_t)n * HW + hw0;
    *(float4*)(feaS + o) = fS; *(float4*)(feaT + o) = fT;
    *(float4*)(cmS + o) = cS;  *(float4*)(cmT + o) = cT;
}

// ---------------- K1b: per-channel sums ----------------
// grid (256, 8): blockIdx.x = c, blockIdx.y = n
__global__ __launch_bounds__(256) void k_pass1b(
    const float* __restrict__ S, const float* __restrict__ T,
    float* __restrict__ chS, float* __restrict__ chT)
{
    __shared__ float red[256];
    int n = blockIdx.y, c = blockIdx.x;
    size_t base = ((size_t)n * CCH + c) * HW;
    float sS = 0.f, sT = 0.f;
    for (int i = 0; i < 16; ++i) {
        size_t o = base + i * 1024 + threadIdx.x * 4;
        float4 s = *(const float4*)(S + o);
        float4 t = *(const float4*)(T + o);
        sS += fabsf(s.x) + fabsf(s.y) + fabsf(s.z) + fabsf(s.w);
        sT += fabsf(t.x) + fabsf(t.y) + fabsf(t.z) + fabsf(t.w);
    }
    float rS = block_reduce_sum(sS, red);
    float rT = block_reduce_sum(sT, red);
    if (threadIdx.x == 0) { chS[n * CCH + c] = rS; chT[n * CCH + c] = rT; }
}

// ---------------- K2: spatial softmax params for feaS,feaT,cmS,cmT ----------------
// grid 32: a = blockIdx.x >> 3, n = blockIdx.x & 7
__global__ __launch_bounds__(256) void k_smax(const float* __restrict__ ws0, float* __restrict__ par) {
    __shared__ float red[256];
    int a = blockIdx.x >> 3, n = blockIdx.x & 7;
    const float* p = ws0 + (size_t)a * 131072u + (size_t)n * HW;
    float sc = (a < 2) ? (1.f / 128.f) : 1.f;
    float mx = -3.4e38f;
    for (int i = 0; i < 64; ++i) mx = fmaxf(mx, p[i * 256 + threadIdx.x] * sc);
    mx = block_reduce_max(mx, red);
    float se = 0.f;
    for (int i = 0; i < 64; ++i) se += __expf(p[i * 256 + threadIdx.x] * sc - mx);
    se = block_reduce_sum(se, red);
    if (threadIdx.x == 0) { par[(a * 8 + n) * 2] = mx; par[(a * 8 + n) * 2 + 1] = se; }
}

// ---------------- K3: masks, spatial attentions, weights, spatial L1 ----------------
// grid 8 (n)
__global__ __launch_bounds__(256) void k_masks(
    const float* __restrict__ gt,
    const float* __restrict__ feaS, const float* __restrict__ feaT,
    const float* __restrict__ cmS, const float* __restrict__ cmT,
    const float* __restrict__ par,
    float* __restrict__ smS, float* __restrict__ smT,
    float* __restrict__ wfg, float* __restrict__ wbg, float* __restrict__ acc)
{
    __shared__ float red[256];
    __shared__ float bx0[NB], bx1[NB], by0[NB], by1[NB], bar[NB];
    __shared__ float s_sbg;
    int n = blockIdx.x, tid = threadIdx.x;
    if (tid < NB) {
        const float* b = gt + ((size_t)n * NB + tid) * 4;
        float wmin = floorf(b[0] * 0.125f), hmin = floorf(b[1] * 0.125f);
        float wmax = ceilf(b[2] * 0.125f),  hmax = ceilf(b[3] * 0.125f);
        bx0[tid] = wmin; bx1[tid] = wmax; by0[tid] = hmin; by1[tid] = hmax;
        bar[tid] = 1.f / ((hmax + 1.f - hmin) * (wmax + 1.f - wmin));
    }
    __syncthreads();
    // phase 1: fg mask + bg count
    float bgc = 0.f;
    for (int i = 0; i < 64; ++i) {
        int hw = i * 256 + tid;
        float h = (float)(hw >> 7), w = (float)(hw & 127);
        float fg = 0.f;
        #pragma unroll
        for (int b = 0; b < NB; ++b) {
            bool in = (h >= by0[b]) & (h <= by1[b]) & (w >= bx0[b]) & (w <= bx1[b]);
            fg = fmaxf(fg, in ? bar[b] : 0.f);
        }
        wfg[(size_t)n * HW + hw] = fg;
        bgc += (fg > 0.f) ? 0.f : 1.f;
    }
    float sbg = block_reduce_sum(bgc, red);
    if (tid == 0) s_sbg = sbg;
    __syncthreads();
    float invs = (s_sbg > 0.f) ? (1.f / s_sbg) : 1.f;
    float mxFS = par[(0 * 8 + n) * 2], seFS = par[(0 * 8 + n) * 2 + 1];
    float mxFT = par[(1 * 8 + n) * 2], seFT = par[(1 * 8 + n) * 2 + 1];
    float mxCS = par[(2 * 8 + n) * 2], seCS = par[(2 * 8 + n) * 2 + 1];
    float mxCT = par[(3 * 8 + n) * 2], seCT = par[(3 * 8 + n) * 2 + 1];
    // phase 2: attentions + weights + spatial L1
    float l1 = 0.f;
    for (int i = 0; i < 64; ++i) {
        size_t o = (size_t)n * HW + i * 256 + tid;
        float Ss = 16384.f * __expf(feaS[o] * (1.f / 128.f) - mxFS) / seFS;
        float St = 16384.f * __expf(feaT[o] * (1.f / 128.f) - mxFT) / seFT;
        l1 += fabsf(Ss - St);
        smS[o] = __expf(cmS[o] - mxCS) / seCS;
        smT[o] = __expf(cmT[o] - mxCT) / seCT;
        float fg = wfg[o];
        wfg[o] = St * fg;
        wbg[o] = St * ((fg > 0.f) ? 0.f : invs);
    }
    l1 = block_reduce_sum(l1, red);
    if (tid == 0) atomicAdd(&acc[3], l1);
}

// ---------------- K4: channel softmax + channel L1 ----------------
// grid 8 (n), tid = c
__global__ __launch_bounds__(256) void k_chan(
    const float* __restrict__ chS, const float* __restrict__ chT,
    float* __restrict__ Cta, float* __restrict__ acc)
{
    __shared__ float red[256];
    int n = blockIdx.x, c = threadIdx.x;
    float lS = chS[n * CCH + c] * (1.f / 8192.f);
    float lT = chT[n * CCH + c] * (1.f / 8192.f);
    float mxS = block_reduce_max(lS, red);
    float mxT = block_reduce_max(lT, red);
    float eS = __expf(lS - mxS), eT = __expf(lT - mxT);
    float sS = block_reduce_sum(eS, red);
    float sT = block_reduce_sum(eT, red);
    float Cs = 256.f * eS / sS, Ct = 256.f * eT / sT;
    Cta[n * CCH + c] = Ct;
    float l1 = block_reduce_sum(fabsf(Cs - Ct), red);
    if (c == 0) atomicAdd(&acc[2], l1);
}

// ---------------- K5: big pass 2 ----------------
// grid (256, 8): blockIdx.x = c, blockIdx.y = n
__global__ __launch_bounds__(256) void k_pass2(
    const float* __restrict__ S, const float* __restrict__ T,
    const float* __restrict__ smS, const float* __restrict__ smT,
    const float* __restrict__ wfg, const float* __restrict__ wbg,
    const float* __restrict__ Cta,
    float* __restrict__ ctxS, float* __restrict__ ctxT,
    float* __restrict__ SD, float* __restrict__ SS2, float* __restrict__ acc)
{
    __shared__ float part[8][6];
    int n = blockIdx.y, c = blockIdx.x;
    int tid = threadIdx.x, wv = tid >> 5, lane = tid & 31;
    size_t base = ((size_t)n * CCH + c) * HW;
    size_t abase = (size_t)n * HW;
    float cs = 0.f, ct = 0.f, sd = 0.f, ss2 = 0.f, fgp = 0.f, bgp = 0.f;
    for (int i = 0; i < 16; ++i) {
        size_t off = i * 1024 + tid * 4;
        float4 s = *(const float4*)(S + base + off);
        float4 t = *(const float4*)(T + base + off);
        float4 ms = *(const float4*)(smS + abase + off);
        float4 mt = *(const float4*)(smT + abase + off);
        float4 wf = *(const float4*)(wfg + abase + off);
        float4 wb = *(const float4*)(wbg + abase + off);
        cs += s.x*ms.x + s.y*ms.y + s.z*ms.z + s.w*ms.w;
        ct += t.x*mt.x + t.y*mt.y + t.z*mt.z + t.w*mt.w;
        float dx = s.x - t.x, dy = s.y - t.y, dz = s.z - t.z, dw = s.w - t.w;
        sd  += dx + dy + dz + dw;
        ss2 += dx*dx + dy*dy + dz*dz + dw*dw;
        fgp += dx*dx*wf.x + dy*dy*wf.y + dz*dz*wf.z + dw*dw*wf.w;
        bgp += dx*dx*wb.x + dy*dy*wb.y + dz*dz*wb.z + dw*dw*wb.w;
    }
    // wave-level reduce 6 values
    #pragma unroll
    for (int off = 16; off; off >>= 1) {
        cs  += __shfl_xor(cs, off);  ct  += __shfl_xor(ct, off);
        sd  += __shfl_xor(sd, off);  ss2 += __shfl_xor(ss2, off);
        fgp += __shfl_xor(fgp, off); bgp += __shfl_xor(bgp, off);
    }
    if (lane == 0) {
        part[wv][0] = cs; part[wv][1] = ct; part[wv][2] = sd;
        part[wv][3] = ss2; part[wv][4] = fgp; part[wv][5] = bgp;
    }
    __syncthreads();
    if (tid < 6) {
        float v = 0.f;
        #pragma unroll
        for (int w = 0; w < 8; ++w) v += part[w][tid];
        int idx = n * CCH + c;
        float ctw = Cta[idx];
        if (tid == 0) ctxS[idx] = v;
        else if (tid == 1) ctxT[idx] = v;
        else if (tid == 2) SD[idx] = v;
        else if (tid == 3) SS2[idx] = v;
        else if (tid == 4) atomicAdd(&acc[0], ctw * v);
        else atomicAdd(&acc[1], ctw * v);
    }
}

// ---------------- K7: GC-block MLPs via fp32 WMMA + final scalar ----------------
// 1 block, 256 threads (8 waves). GEMM1: [16,256]x[256,128]; GEMM2: [16,128]x[128,256]
__global__ __launch_bounds__(256) void k_mlp_final(
    const float* __restrict__ ctxS, const float* __restrict__ ctxT,
    const float* __restrict__ w1s, const float* __restrict__ b1s,
    const float* __restrict__ gs,  const float* __restrict__ bes,
    const float* __restrict__ w2s, const float* __restrict__ b2s,
    const float* __restrict__ w1t, const float* __restrict__ b1t,
    const float* __restrict__ gt,  const float* __restrict__ bet,
    const float* __restrict__ w2t, const float* __restrict__ b2t,
    const float* __restrict__ SD, const float* __restrict__ SS2,
    const float* __restrict__ acc, float* __restrict__ out)
{
    __shared__ float ctxA[16 * 257];   // padded: conflict-reduced transposed reads
    __shared__ float yb[16 * 132];
    __shared__ float yn[16 * 132];
    __shared__ float addD[2 * 8 * 256];
    __shared__ float red[256];
    int tid = threadIdx.x, wv = tid >> 5, lane = tid & 31;
    int m16 = lane & 15, kh = lane >> 4;

    for (int i = tid; i < 16 * 132; i += 256) yn[i] = 0.f;  // rows 8..15 stay zero

    for (int br = 0; br < 2; ++br) {
        const float* ctx = br ? ctxT : ctxS;
        const float* w1 = br ? w1t : w1s; const float* b1 = br ? b1t : b1s;
        const float* g  = br ? gt  : gs;  const float* be = br ? bet : bes;
        const float* w2 = br ? w2t : w2s; const float* b2 = br ? b2t : b2s;
        __syncthreads();
        for (int i = tid; i < 16 * 256; i += 256) {
            int r = i >> 8, col = i & 255;
            ctxA[r * 257 + col] = (r < 8) ? ctx[r * 256 + col] : 0.f;
        }
        __syncthreads();
        // ---- GEMM1: y = ctx @ w1^T ; wave wv -> column tile jt = wv
        {
            v8f accv = {0.f,0.f,0.f,0.f,0.f,0.f,0.f,0.f};
            int j = (wv << 4) + m16;
            for (int kb = 0; kb < 64; ++kb) {
                int k0 = (kb << 2) + (kh << 1);
                v2f a, b;
                a.x = ctxA[m16 * 257 + k0]; a.y = ctxA[m16 * 257 + k0 + 1];
                b.x = w1[j * 256 + k0];     b.y = w1[j * 256 + k0 + 1];
                accv = __builtin_amdgcn_wmma_f32_16x16x4_f32(
                    false, a, false, b, (short)0, accv, false, false);
            }
            float bj = b1[j];
            #pragma unroll
            for (int r = 0; r < 8; ++r) {
                int row = (lane < 16) ? r : (r + 8);
                yb[row * 132 + j] = accv[r] + bj;
            }
        }
        __syncthreads();
        // ---- LayerNorm + ReLU over 128 features; wave wv owns row wv (rows 0..7)
        {
            float vals[4]; float s0 = 0.f, s1 = 0.f;
            #pragma unroll
            for (int q = 0; q < 4; ++q) {
                vals[q] = yb[wv * 132 + lane * 4 + q];
                s0 += vals[q]; s1 += vals[q] * vals[q];
            }
            #pragma unroll
            for (int off = 16; off; off >>= 1) { s0 += __shfl_xor(s0, off); s1 += __shfl_xor(s1, off); }
            float mean = s0 * (1.f / 128.f);
            float var  = s1 * (1.f / 128.f) - mean * mean;
            float rstd = rsqrtf(var + 1e-5f);
            #pragma unroll
            for (int q = 0; q < 4; ++q) {
                int jj = lane * 4 + q;
                float v = (vals[q] - mean) * rstd * g[jj] + be[jj];
                yn[wv * 132 + jj] = fmaxf(v, 0.f);
            }
        }
        __syncthreads();
        // ---- GEMM2: add = relu(yn) @ w2^T ; wave wv -> tiles wv, wv+8
        for (int half = 0; half < 2; ++half) {
            int jt = wv + half * 8;
            int j = (jt << 4) + m16;
            v8f accv = {0.f,0.f,0.f,0.f,0.f,0.f,0.f,0.f};
            for (int kb = 0; kb < 32; ++kb) {
                int k0 = (kb << 2) + (kh << 1);
                v2f a, b;
                a.x = yn[m16 * 132 + k0]; a.y = yn[m16 * 132 + k0 + 1];
                b.x = w2[j * 128 + k0];   b.y = w2[j * 128 + k0 + 1];
                accv = __builtin_amdgcn_wmma_f32_16x16x4_f32(
                    false, a, false, b, (short)0, accv, false, false);
            }
            float bj = b2[j];
            if (lane < 16) {
                #pragma unroll
                for (int r = 0; r < 8; ++r) addD[br * 2048 + r * 256 + j] = accv[r] + bj;
            }
        }
        __syncthreads();
    }
    // ---- rela loss: sum SS2 + 2*a*SD + HW*a^2, a = add_s - add_t
    float rela = 0.f;
    for (int i = tid; i < 2048; i += 256) {
        float a = addD[i] - addD[2048 + i];
        rela += SS2[i] + 2.f * a * SD[i] + 16384.f * a * a;
    }
    red[tid] = rela; __syncthreads();
    for (int s = 128; s; s >>= 1) { if (tid < s) red[tid] += red[tid + s]; __syncthreads(); }
    if (tid == 0) {
        float fg = acc[0], bg = acc[1], chl1 = acc[2], spl1 = acc[3];
        out[0] = (1e-3f * fg + 5e-4f * bg + 1e-3f * (chl1 + spl1) + 5e-6f * red[0]) * (1.f / 8.f);
    }
}

extern "C" void kernel_launch(void* const* d_in, const int* in_sizes, int n_in,
                              void* d_out, int out_size, void* d_ws, size_t ws_size,
                              hipStream_t stream) {
    const float* S    = (const float*)d_in[0];
    const float* T    = (const float*)d_in[1];
    const float* gt   = (const float*)d_in[2];
    const float* wms  = (const float*)d_in[3];
    const float* wmt  = (const float*)d_in[5];
    const float* w1s  = (const float*)d_in[7];
    const float* b1s  = (const float*)d_in[8];
    const float* gs   = (const float*)d_in[9];
    const float* bes  = (const float*)d_in[10];
    const float* w2s  = (const float*)d_in[11];
    const float* b2s  = (const float*)d_in[12];
    const float* w1t  = (const float*)d_in[13];
    const float* b1t  = (const float*)d_in[14];
    const float* gtt  = (const float*)d_in[15];
    const float* bet  = (const float*)d_in[16];
    const float* w2t  = (const float*)d_in[17];
    const float* b2t  = (const float*)d_in[18];

    float* ws = (float*)d_ws;
    float* feaS = ws + OFF_FEA_S; float* feaT = ws + OFF_FEA_T;
    float* cmS  = ws + OFF_CM_S;  float* cmT  = ws + OFF_CM_T;
    float* smS  = ws + OFF_SM_S;  float* smT  = ws + OFF_SM_T;
    float* wfg  = ws + OFF_WFG;   float* wbg  = ws + OFF_WBG;
    float* chS  = ws + OFF_CH_S;  float* chT  = ws + OFF_CH_T;
    float* Cta  = ws + OFF_CTA;
    float* ctxS = ws + OFF_CTX_S; float* ctxT = ws + OFF_CTX_T;
    float* SDp  = ws + OFF_SD;    float* SS2p = ws + OFF_SS2;
    float* par  = ws + OFF_SPAR;  float* acc  = ws + OFF_ACC;
    float* out  = (float*)d_out;

    k_init<<<1, 64, 0, stream>>>(acc);
    k_pass1a<<<dim3(16, NBATCH), 256, 0, stream>>>(S, T, wms, wmt, feaS, feaT, cmS, cmT);
    k_pass1b<<<dim3(CCH, NBATCH), 256, 0, stream>>>(S, T, chS, chT);
    k_smax<<<32, 256, 0, stream>>>(ws, par);
    k_masks<<<NBATCH, 256, 0, stream>>>(gt, feaS, feaT, cmS, cmT, par, smS, smT, wfg, wbg, acc);
    k_chan<<<NBATCH, 256, 0, stream>>>(chS, chT, Cta, acc);
    k_pass2<<<dim3(CCH, NBATCH), 256, 0, stream>>>(S, T, smS, smT, wfg, wbg, Cta,
                                                   ctxS, ctxT, SDp, SS2p, acc);
    k_mlp_final<<<1, 256, 0, stream>>>(ctxS, ctxT,
                                       w1s, b1s, gs, bes, w2s, b2s,
                                       w1t, b1t, gtt, bet, w2t, b2t,
                                       SDp, SS2p, acc, out);
}